// IsoneutralMixing_39505109189235
// MI455X (gfx1250) — compile-verified
//
#include <hip/hip_runtime.h>
#include <math.h>

// Isoneutral mixing (VEROS) for MI455X / gfx1250.
//
// Roofline: 304 MB compulsory stores + ~200 MB compulsory reads -> ~22 us at
// 23.3 TB/s HBM. Store-bandwidth-bound stencil; no matmul structure -> WMMA
// inapplicable. CDNA5 levers used:
//   * GLOBAL_LOAD_ASYNC_TO_LDS_B32 staging of the strided (stride-12B)
//     temp/salt TAU=0 stencil tile into LDS (ASYNCcnt + s_wait_asynccnt),
//     de-duplicating ~26 strided gathers/thread into one tile fetch/block
//   * z-contiguous wave32 layout (128B coalesced bursts)
//   * b128 non-temporal stores for the 304MB of streaming output
//   * global_prefetch_b8 run-ahead on the x+2 plane

#define XD 256
#define YD 256
#define ZD 64

typedef __attribute__((ext_vector_type(4))) float v4f;

__device__ __forceinline__ float l3(const float* __restrict__ a, int x, int y, int z) {
  return a[(x * YD + y) * ZD + z];
}

__device__ __forceinline__ float dm_taper(float s) {
  return 0.5f * (1.0f + tanhf((-fabsf(s) + 0.001f) / 0.001f));
}

// rho derivative w.r.t. T from the T value itself (mask applied by caller)
__device__ __forceinline__ float drho_dT(float Tval, float ztz) {
  const float cg = (float)(1.1e-8 * 9.81 * 1024.0);   // gammas*grav*rho0
  const float theta0 = (float)(283.0 - 273.15);
  return -(1.0e-5f * (Tval - theta0) + 1.67e-4f * (1.0f - cg * ztz)) * 1024.0f;
}

__global__ __launch_bounds__(256) void iso_kernel(
    const float* __restrict__ maskT, const float* __restrict__ maskU,
    const float* __restrict__ maskV, const float* __restrict__ maskW,
    const float* __restrict__ dxt,  const float* __restrict__ dxu,
    const float* __restrict__ dyt,  const float* __restrict__ dyu,
    const float* __restrict__ dzt,  const float* __restrict__ dzw,
    const float* __restrict__ cost, const float* __restrict__ cosu,
    const float* __restrict__ salt, const float* __restrict__ temp,
    const float* __restrict__ zt,   const float* __restrict__ Kiso,
    float* __restrict__ K11, float* __restrict__ K22, float* __restrict__ K33,
    float* __restrict__ Aez, float* __restrict__ Anz,
    float* __restrict__ Abx, float* __restrict__ Aby)
{
  const int z  = threadIdx.x;              // 0..63 (contiguous)
  const int ty = threadIdx.y;              // 0..3
  const int y0 = blockIdx.y * 4;
  const int y  = y0 + ty;                  // 0..255
  const int x  = blockIdx.x;               // 0..255
  const int idx3 = (x * YD + y) * ZD + z;
  const int idx5 = idx3 * 4;

  // LDS tile of the TAU=0 temp/salt slices covering the stencil halo:
  // x-1..x+1 (px 0..2) times y0-1..y0+4 (py 0..5) times full z.
  __shared__ float tT[3 * 6 * ZD];
  __shared__ float tS[3 * 6 * ZD];

  // ---- async stage: 2304 elements, 9 per thread, per-lane stride-3 gather ----
  const int tid = ty * ZD + z;             // 0..255
#pragma unroll
  for (int k = 0; k < 9; ++k) {
    int l = tid + k * 256;                 // 0..2303
    int isS = (l >= 1152) ? 1 : 0;
    int e = l - isS * 1152;                // 0..1151
    const float* src = isS ? salt : temp;
    float* dst = isS ? tS : tT;
    int row = e >> 6;                      // 0..17
    int zz = e & 63;
    int px = row / 6;
    int py = row - px * 6;
    int gx = min(max(x - 1 + px, 0), XD - 1);   // clamped halo rows are unused
    int gy = min(max(y0 - 1 + py, 0), YD - 1);
    const float* g = &src[((gx * YD + gy) * ZD + zz) * 3];
    // low 32 bits of a generic LDS pointer == workgroup-relative LDS address
    unsigned ldsaddr = (unsigned)(unsigned long long)&dst[e];
    unsigned long long ga = (unsigned long long)g;
    asm volatile("global_load_async_to_lds_b32 %0, %1, off"
                 :: "v"(ldsaddr), "v"(ga) : "memory");
  }
  asm volatile("s_wait_asynccnt 0x0" ::: "memory");
  __syncthreads();

  // Run-ahead prefetch of the x+2 temp plane (gfx1250 global_prefetch_b8).
  if (x + 2 < XD) {
    __builtin_prefetch(&temp[(((x + 2) * YD + y) * ZD + z) * 3], 0, 1);
    __builtin_prefetch(&salt[(((x + 2) * YD + y) * ZD + z) * 3], 0, 1);
  }

  // Tile accessors: offsets relative to this thread's (x, y).
  auto TL = [&](int dx, int dy, int zz) -> float {
    return tT[((dx + 1) * 6 + (ty + 1 + dy)) * ZD + zz];
  };
  auto SL = [&](int dx, int dy, int zz) -> float {
    return tS[((dx + 1) * 6 + (ty + 1 + dy)) * ZD + zz];
  };
  // derivatives (zero-padded per the reference jnp.pad at the top z face)
  auto dTdz_f = [&](int dx, int dy, int zz) -> float {
    if (zz >= ZD - 1) return 0.0f;
    return l3(maskW, x + dx, y + dy, zz) * (TL(dx, dy, zz + 1) - TL(dx, dy, zz)) / dzw[zz];
  };
  auto dSdz_f = [&](int dx, int dy, int zz) -> float {
    if (zz >= ZD - 1) return 0.0f;
    return l3(maskW, x + dx, y + dy, zz) * (SL(dx, dy, zz + 1) - SL(dx, dy, zz)) / dzw[zz];
  };
  auto dTdx_f = [&](int dx, int zz) -> float {   // always dy==0 at call sites
    return l3(maskU, x + dx, y, zz) * (TL(dx + 1, 0, zz) - TL(dx, 0, zz)) / (dxu[x + dx] * cost[y]);
  };
  auto dSdx_f = [&](int dx, int zz) -> float {
    return l3(maskU, x + dx, y, zz) * (SL(dx + 1, 0, zz) - SL(dx, 0, zz)) / (dxu[x + dx] * cost[y]);
  };
  auto dTdy_f = [&](int dy, int zz) -> float {   // always dx==0 at call sites
    return l3(maskV, x, y + dy, zz) * (TL(0, dy + 1, zz) - TL(0, dy, zz)) / dyu[y + dy];
  };
  auto dSdy_f = [&](int dy, int zz) -> float {
    return l3(maskV, x, y + dy, zz) * (SL(0, dy + 1, zz) - SL(0, dy, zz)) / dyu[y + dy];
  };

  const float DRDS = (float)(0.00078 * 1024.0);
  const float EPS = 1e-20f;

  float k11 = 0.0f, k22 = 0.0f, k33 = 0.0f;
  float ae[4]  = {0.f, 0.f, 0.f, 0.f};
  float an[4]  = {0.f, 0.f, 0.f, 0.f};
  float abx[4] = {0.f, 0.f, 0.f, 0.f};
  float aby[4] = {0.f, 0.f, 0.f, 0.f};

  const bool in1 = (x >= 1 && x <= XD - 3 && y >= 2 && y <= YD - 3);
  const bool in2 = (x >= 2 && x <= XD - 3 && y >= 1 && y <= YD - 3);
  const bool in3 = (x >= 2 && x <= XD - 3 && y >= 2 && y <= YD - 3 && z <= ZD - 2);

  // ---------------- Part 1: K_11 / Ai_ez (east faces) ----------------
  if (in1) {
    const float dTdx_c = dTdx_f(0, z);
    const float dSdx_c = dSdx_f(0, z);
    const float mU = l3(maskU, x, y, z);
    float diffloc;
    if (z == 0)
      diffloc = 0.5f * (l3(Kiso, x, y, 0) + l3(Kiso, x + 1, y, 0));
    else
      diffloc = 0.25f * (l3(Kiso, x, y, z) + l3(Kiso, x, y, z - 1) +
                         l3(Kiso, x + 1, y, z) + l3(Kiso, x + 1, y, z - 1));
    float sumz = 0.0f;
#pragma unroll
    for (int kr = 0; kr < 2; ++kr) {
      const int ki = 1 - kr;
      if (z < ki) continue;
      const int zk = z - ki;
      const float dzwk = dzw[zk];
#pragma unroll
      for (int ip = 0; ip < 2; ++ip) {
        const float mT = l3(maskT, x + ip, y, z);
        const float rT = drho_dT(TL(ip, 0, z), zt[z]) * mT;
        const float rS = DRDS * mT;
        const float drodxe = rT * dTdx_c + rS * dSdx_c;
        const float drodze = rT * dTdz_f(ip, 0, zk) + rS * dSdz_f(ip, 0, zk);
        const float sxe = -drodxe / (fminf(0.0f, drodze) - EPS);
        const float taper = dm_taper(sxe);
        sumz += dzwk * mU * fmaxf(50.0f, diffloc * taper);
        ae[ip * 2 + kr] = taper * sxe * mU;
      }
    }
    k11 = sumz / (4.0f * dzt[z]);
  }

  // ---------------- Part 2: K_22 / Ai_nz (north faces) ----------------
  if (in2) {
    const float dTdy_c = dTdy_f(0, z);
    const float dSdy_c = dSdy_f(0, z);
    const float mV = l3(maskV, x, y, z);
    float diffloc;
    if (z == 0)
      diffloc = 0.5f * (l3(Kiso, x, y, 0) + l3(Kiso, x, y + 1, 0));
    else
      diffloc = 0.25f * (l3(Kiso, x, y, z) + l3(Kiso, x, y, z - 1) +
                         l3(Kiso, x, y + 1, z) + l3(Kiso, x, y + 1, z - 1));
    float sumz = 0.0f;
#pragma unroll
    for (int kr = 0; kr < 2; ++kr) {
      const int ki = 1 - kr;
      if (z < ki) continue;
      const int zk = z - ki;
      const float dzwk = dzw[zk];
#pragma unroll
      for (int jp = 0; jp < 2; ++jp) {
        const float mT = l3(maskT, x, y + jp, z);
        const float rT = drho_dT(TL(0, jp, z), zt[z]) * mT;
        const float rS = DRDS * mT;
        const float drodyn = rT * dTdy_c + rS * dSdy_c;
        const float drodzn = rT * dTdz_f(0, jp, zk) + rS * dSdz_f(0, jp, zk);
        const float syn = -drodyn / (fminf(0.0f, drodzn) - EPS);
        const float taper = dm_taper(syn);
        sumz += dzwk * mV * fmaxf(50.0f, diffloc * taper);
        an[jp * 2 + kr] = taper * syn * mV;
      }
    }
    k22 = sumz / (4.0f * dzt[z]);
  }

  // ---------------- Part 3: K_33 / Ai_bx / Ai_by (bottom faces) --------
  if (in3) {
    const float mW = l3(maskW, x, y, z);
    const float Kc = l3(Kiso, x, y, z);
    const float dTdz_c = dTdz_f(0, 0, z);
    const float dSdz_c = dSdz_f(0, 0, z);
    float sumx = 0.0f, sumy = 0.0f;
#pragma unroll
    for (int kr = 0; kr < 2; ++kr) {
      const int zr = z + kr;
      const float mT = l3(maskT, x, y, zr);
      const float rT = drho_dT(TL(0, 0, zr), zt[zr]) * mT;
      const float rS = DRDS * mT;
      const float drodzb = rT * dTdz_c + rS * dSdz_c;
      const float denom = fminf(0.0f, drodzb) - EPS;
#pragma unroll
      for (int ip = 0; ip < 2; ++ip) {
        const int dx = ip - 1;                       // xs = x-1+ip
        const float drodxb = rT * dTdx_f(dx, zr) + rS * dSdx_f(dx, zr);
        const float sxb = -drodxb / denom;
        const float taper = dm_taper(sxb);
        sumx += dxu[x + dx] * Kc * taper * sxb * sxb * mW;
        abx[ip * 2 + kr] = taper * sxb * mW;
      }
#pragma unroll
      for (int jp = 0; jp < 2; ++jp) {
        const int dy = jp - 1;                       // ys = y-1+jp
        const float facty = cosu[y + dy] * dyu[y + dy];
        const float drodyb = rT * dTdy_f(dy, zr) + rS * dSdy_f(dy, zr);
        const float syb = -drodyb / denom;
        const float taper = dm_taper(syb);
        sumy += facty * Kc * taper * syb * syb * mW;
        aby[jp * 2 + kr] = taper * syb * mW;
      }
    }
    k33 = sumx / (4.0f * dxt[x]) + sumy / (4.0f * dyt[y] * cost[y]);
  }
  // z == ZD-1 and out-of-region stay 0 (matches reference zeros / explicit set)

  // ---------------- Streaming stores (non-temporal, b128-packed) -------
  __builtin_nontemporal_store(k11, K11 + idx3);
  __builtin_nontemporal_store(k22, K22 + idx3);
  __builtin_nontemporal_store(k33, K33 + idx3);
  v4f vez = {ae[0], ae[1], ae[2], ae[3]};
  v4f vnz = {an[0], an[1], an[2], an[3]};
  v4f vbx = {abx[0], abx[1], abx[2], abx[3]};
  v4f vby = {aby[0], aby[1], aby[2], aby[3]};
  __builtin_nontemporal_store(vez, (v4f*)(Aez + idx5));
  __builtin_nontemporal_store(vnz, (v4f*)(Anz + idx5));
  __builtin_nontemporal_store(vbx, (v4f*)(Abx + idx5));
  __builtin_nontemporal_store(vby, (v4f*)(Aby + idx5));
}

extern "C" void kernel_launch(void* const* d_in, const int* in_sizes, int n_in,
                              void* d_out, int out_size, void* d_ws, size_t ws_size,
                              hipStream_t stream) {
  (void)in_sizes; (void)n_in; (void)out_size; (void)d_ws; (void)ws_size;
  const float* maskT = (const float*)d_in[0];
  const float* maskU = (const float*)d_in[1];
  const float* maskV = (const float*)d_in[2];
  const float* maskW = (const float*)d_in[3];
  const float* dxt   = (const float*)d_in[4];
  const float* dxu   = (const float*)d_in[5];
  const float* dyt   = (const float*)d_in[6];
  const float* dyu   = (const float*)d_in[7];
  const float* dzt   = (const float*)d_in[8];
  const float* dzw   = (const float*)d_in[9];
  const float* cost  = (const float*)d_in[10];
  const float* cosu  = (const float*)d_in[11];
  const float* salt  = (const float*)d_in[12];
  const float* temp  = (const float*)d_in[13];
  const float* zt    = (const float*)d_in[14];
  const float* Kiso  = (const float*)d_in[15];
  // d_in[16..22] are the zero-initialized output seeds; we write all elements.

  float* out = (float*)d_out;
  const int N3 = XD * YD * ZD;
  float* K11 = out;
  float* K22 = out + (size_t)N3;
  float* K33 = out + (size_t)2 * N3;
  float* Aez = out + (size_t)3 * N3;
  float* Anz = Aez + (size_t)4 * N3;
  float* Abx = Anz + (size_t)4 * N3;
  float* Aby = Abx + (size_t)4 * N3;

  dim3 block(ZD, 4, 1);          // 256 threads = 8 wave32; z contiguous
  dim3 grid(XD, YD / 4, 1);
  iso_kernel<<<grid, block, 0, stream>>>(maskT, maskU, maskV, maskW,
                                         dxt, dxu, dyt, dyu, dzt, dzw,
                                         cost, cosu, salt, temp, zt, Kiso,
                                         K11, K22, K33, Aez, Anz, Abx, Aby);
}